// Embedder_68109591380517
// MI455X (gfx1250) — compile-verified
//
#include <hip/hip_runtime.h>

// ---------------------------------------------------------------------------
// Embedder segment-mean as a one-hot WMMA matmul (gfx1250 / CDNA5, wave32).
//
//   per batch b:  sums(48x128) = A(48x2048) x E(2048x128)
//     A[t,l] = exp(embW[id_l]) if clamp(floor(T_l),0,47)==t else 0
//     E[l,:] = embX[id_l]
//   out[b,t,:] = sums[t,:] / (count[t] + 1e-6)
//
// One workgroup per batch, 256 threads = 8 waves. Wave w owns N-columns
// [16w,16w+16) and 3 M-tiles (48 rows) of fp32 WMMA accumulators.
// K=2048 processed in chunks of 64 via V_WMMA_F32_16X16X4_F32.
// ---------------------------------------------------------------------------

typedef __attribute__((ext_vector_type(2))) float v2f;
typedef __attribute__((ext_vector_type(8))) float v8f;

#define BB    256   // batches
#define LL    2048  // events per batch
#define DD    128   // embedding dim
#define TMAX  48    // output bins
#define KC    64    // K-chunk
#define AP    66    // A tile row stride (KC + 2 pad, even -> 8B-aligned b64 loads)
#define BP    132   // B tile row stride (DD + 4 pad, keeps float4 stores 16B-aligned)

__global__ __launch_bounds__(256) void embedder_wmma_kernel(
    const float* __restrict__ X,      // (BB, LL, 2) : [T, id_as_float]
    const float* __restrict__ embX,   // (V, DD)
    const float* __restrict__ embW,   // (V+1, 1)
    float* __restrict__ out)          // (BB, TMAX, DD)
{
    __shared__ float lds_w [LL];        //  8 KB  exp(embW[id])
    __shared__ int   lds_pk[LL];        //  8 KB  id | (bin<<16)
    __shared__ float lds_cnt[TMAX];     //  bin counts
    __shared__ float ldsA[TMAX * AP];   // 12.4 KB  A tile (48 x KC, row-major)
    __shared__ float ldsB[KC * BP];     // 33.8 KB  B tile (KC x 128, row-major)

    const int t    = threadIdx.x;
    const int b    = blockIdx.x;
    const int wv   = t >> 5;            // wave id 0..7  -> N-tile
    const int lane = t & 31;
    const int half = lane >> 4;         // 0: K pair {0,1}, 1: K pair {2,3}
    const int l16  = lane & 15;

    // ---------------- Phase 1: decode events, bin counts -------------------
    if (t < TMAX) lds_cnt[t] = 0.0f;
    __syncthreads();

    const float2* Xp = (const float2*)X;
#pragma unroll
    for (int i = 0; i < LL / 256; ++i) {
        int l = t + 256 * i;                       // coalesced
        float2 x = Xp[b * LL + l];
        int id  = (int)x.y;
        int bin = (int)floorf(x.x);
        bin = min(max(bin, 0), TMAX - 1);
        lds_w[l]  = __expf(embW[id]);
        lds_pk[l] = id | (bin << 16);
        atomicAdd(&lds_cnt[bin], 1.0f);            // ds_add_f32
    }

    // ---------------- Phase 2: K-loop of one-hot WMMA matmuls --------------
    v8f acc[3] = {};                               // 3 M-tiles x (16x16 f32 C)

#pragma unroll 1
    for (int ch = 0; ch < LL / KC; ++ch) {
        __syncthreads();   // phase1 done / previous chunk's WMMA reads done

        // Build A tile: thread t<KC owns column t (zero it, then drop w at bin).
        if (t < KC) {
            int   k   = ch * KC + t;
            int   pk  = lds_pk[k];
            float w   = lds_w[k];
            int   bin = pk >> 16;
#pragma unroll
            for (int r = 0; r < TMAX; ++r) ldsA[r * AP + t] = 0.0f;
            ldsA[bin * AP + t] = w;
        }

        // Build B tile: gather 64 embX rows (512B each), coalesced float4.
#pragma unroll
        for (int i = 0; i < (KC * DD / 4) / 256; ++i) {   // 8 float4 / thread
            int idx = t + 256 * i;
            int row = idx >> 5;                    // 0..63 (uniform per wave)
            int c4  = idx & 31;                    // float4 column
            int gid = lds_pk[ch * KC + row] & 0xFFFF;
            float4 v = *(const float4*)(embX + gid * DD + c4 * 4);
            *(float4*)&ldsB[row * BP + c4 * 4] = v;
        }

        __syncthreads();

        // Dense fp32 WMMA over this chunk. EXEC is all-ones here (no branch).
        const int nb = 16 * wv + l16;
#pragma unroll
        for (int k0 = 0; k0 < KC; k0 += 4) {
            int kk = k0 + 2 * half;                // lanes 0-15: K{0,1}; 16-31: K{2,3}
            v2f bf;
            bf.x = ldsB[kk * BP + nb];
            bf.y = ldsB[(kk + 1) * BP + nb];
#pragma unroll
            for (int m = 0; m < 3; ++m) {
                v2f af = *(const v2f*)&ldsA[(m * 16 + l16) * AP + kk];
                acc[m] = __builtin_amdgcn_wmma_f32_16x16x4_f32(
                    /*neg_a=*/false, af, /*neg_b=*/false, bf,
                    /*c_mod=*/(short)0, acc[m],
                    /*reuse_a=*/false, /*reuse_b=*/false);
            }
        }
    }

    // ---------------- Phase 3: normalize + store ---------------------------
    // C layout (32-bit 16x16): lanes 0-15 reg r -> M=r, lanes 16-31 -> M=8+r.
    float* outb = out + b * (TMAX * DD);
    const int n = 16 * wv + l16;
#pragma unroll
    for (int m = 0; m < 3; ++m) {
#pragma unroll
        for (int r = 0; r < 8; ++r) {
            int trow  = m * 16 + half * 8 + r;
            float cnt = lds_cnt[trow] + 1e-6f;     // broadcast LDS read
            outb[trow * DD + n] = acc[m][r] / cnt;
        }
    }
}

extern "C" void kernel_launch(void* const* d_in, const int* in_sizes, int n_in,
                              void* d_out, int out_size, void* d_ws, size_t ws_size,
                              hipStream_t stream) {
    const float* X    = (const float*)d_in[0];   // (256, 2048, 2) f32
    const float* embX = (const float*)d_in[1];   // (6000, 128)   f32
    const float* embW = (const float*)d_in[2];   // (6001, 1)     f32
    float* out = (float*)d_out;                  // (256, 48, 128) f32
    (void)in_sizes; (void)n_in; (void)d_ws; (void)ws_size; (void)out_size;

    embedder_wmma_kernel<<<dim3(BB), dim3(256), 0, stream>>>(X, embX, embW, out);
}